// WassOrderDistance_69028714381920
// MI455X (gfx1250) — compile-verified
//
#include <hip/hip_runtime.h>
#include <math.h>

typedef __attribute__((ext_vector_type(2))) float v2f;
typedef __attribute__((ext_vector_type(8))) float v8f;

static constexpr int NB = 128;   // batches
static constexpr int NN = 256;   // n == m
static constexpr int ND = 128;   // feature dim
static constexpr int NME = NN * NN;           // 65536 elements per batch matrix
#define INV_N   (1.0f / 256.0f)
#define TOLF    0.005f
#define INV_L2  100.0f           // 1/LAMDA2

// ---------------------------------------------------------------- init
__global__ void k_init(float* u, float* v, unsigned* dmaxU, int* done, float* crit) {
    int idx = blockIdx.x * 256 + threadIdx.x;   // grid covers NB*NN = 32768
    if (idx < NB * NN) { u[idx] = INV_N; v[idx] = 0.f; }
    if (idx < NB) dmaxU[idx] = 0u;
    if (idx == 0) { *done = 0; *crit = 0.f; }
}

// ------------------------------------------------- stats (shift) + row norms
__global__ void k_stats(const float* __restrict__ X, const float* __restrict__ Y,
                        float* __restrict__ shiftX, float* __restrict__ shiftY,
                        float* __restrict__ nx, float* __restrict__ ny) {
    __shared__ float sX[ND], sY[ND];
    int b = blockIdx.x, t = threadIdx.x;        // 256 threads
    {
        const float* src = (t < ND) ? X : Y;
        int k = (t < ND) ? t : t - ND;
        const float* base = src + (size_t)b * NN * ND + k;
        float s = 0.f, s2 = 0.f;
        for (int i = 0; i < NN; ++i) { float x = base[(size_t)i * ND]; s += x; s2 = fmaf(x, x, s2); }
        float mean = s * (1.0f / NN);
        float var  = (s2 - s * mean) * (1.0f / (NN - 1));   // unbiased (ddof=1)
        float sh   = mean / sqrtf(var + 1e-4f);             // faithful: mean/sqrt(...)
        if (t < ND) { sX[k] = sh; shiftX[b * ND + k] = sh; }
        else        { sY[k] = sh; shiftY[b * ND + k] = sh; }
    }
    __syncthreads();
    // squared row norms of Xn, Yn (thread t -> row t)
    const float* xr = X + (size_t)b * NN * ND + (size_t)t * ND;
    const float* yr = Y + (size_t)b * NN * ND + (size_t)t * ND;
    float ax = 0.f, ay = 0.f;
    for (int k = 0; k < ND; ++k) {
        float xv = xr[k] - sX[k]; ax = fmaf(xv, xv, ax);
        float yv = yr[k] - sY[k]; ay = fmaf(yv, yv, ay);
    }
    nx[b * NN + t] = ax;
    ny[b * NN + t] = ay;
}

// ------------------------------------------ pairwise sq-dist via f32 WMMA
// block = one 64x64 tile of one batch; 8 waves, 2 WMMA tiles per wave
#define AST 36   // LDS row stride (floats) -> conflict-free fragment gather
__global__ void k_gemm(const float* __restrict__ X, const float* __restrict__ Y,
                       const float* __restrict__ shiftX, const float* __restrict__ shiftY,
                       const float* __restrict__ nx, const float* __restrict__ ny,
                       float* __restrict__ draw, unsigned* __restrict__ dmaxU) {
    __shared__ float As[64 * AST];
    __shared__ float Bs[64 * AST];
    __shared__ float red[256];
    int blk = blockIdx.x;
    int b = blk >> 4;
    int tile = blk & 15;
    int row0 = (tile >> 2) * 64, col0 = (tile & 3) * 64;
    int tid = threadIdx.x;
    int wave = tid >> 5, lane = tid & 31;
    int llo = lane & 15, hi = lane >> 4;

    const v8f zero = {0.f, 0.f, 0.f, 0.f, 0.f, 0.f, 0.f, 0.f};
    v8f acc[2] = {zero, zero};

    const float* Xb = X + (size_t)b * NN * ND;
    const float* Yb = Y + (size_t)b * NN * ND;
    const float* shX = shiftX + b * ND;
    const float* shY = shiftY + b * ND;

    for (int k0 = 0; k0 < ND; k0 += 32) {
        __syncthreads();
        for (int idx = tid; idx < 64 * 32; idx += 256) {
            int r = idx >> 5, c = idx & 31;
            As[r * AST + c] = Xb[(size_t)(row0 + r) * ND + k0 + c] - shX[k0 + c];
            Bs[r * AST + c] = Yb[(size_t)(col0 + r) * ND + k0 + c] - shY[k0 + c];
        }
        __syncthreads();
#pragma unroll
        for (int q = 0; q < 2; ++q) {
            int tt = wave * 2 + q;
            int tr = tt >> 2, tc = tt & 3;
            const float* Arow = &As[(tr * 16 + llo) * AST];
            const float* Brow = &Bs[(tc * 16 + llo) * AST];
            v8f c8 = acc[q];
#pragma unroll
            for (int kk = 0; kk < 32; kk += 4) {
                int kref = kk + 2 * hi;     // lanes 16-31 hold K+2,K+3
                v2f a, bb;
                a.x  = Arow[kref];  a.y  = Arow[kref + 1];
                bb.x = Brow[kref];  bb.y = Brow[kref + 1];
                c8 = __builtin_amdgcn_wmma_f32_16x16x4_f32(
                        false, a, false, bb, (short)0, c8, false, false);
            }
            acc[q] = c8;
        }
    }

    // epilogue: d = |x|^2 + |y|^2 - 2 x.y ; track per-batch max
    float lmax = 0.f;
#pragma unroll
    for (int q = 0; q < 2; ++q) {
        int tt = wave * 2 + q;
        int tr = tt >> 2, tc = tt & 3;
        int j = col0 + tc * 16 + llo;
        float nyj = ny[b * NN + j];
#pragma unroll
        for (int r = 0; r < 8; ++r) {
            int i = row0 + tr * 16 + r + hi * 8;   // C/D layout: VGPR r -> M=r / r+8
            float val = nx[b * NN + i] + nyj - 2.0f * acc[q][r];
            draw[(size_t)b * NME + (size_t)i * NN + j] = val;
            lmax = fmaxf(lmax, val);
        }
    }
    red[tid] = lmax;
    __syncthreads();
    for (int s = 128; s > 0; s >>= 1) {
        if (tid < s) red[tid] = fmaxf(red[tid], red[tid + s]);
        __syncthreads();
    }
    if (tid == 0) atomicMax(&dmaxU[b], __float_as_uint(fmaxf(red[0], 0.f)));
}

// ---------------------------------------- normalize d, build k = p*exp((s-d)/l2)
__global__ void k_expk(float* __restrict__ draw, float* __restrict__ kmat,
                       const unsigned* __restrict__ dmaxU) {
    size_t idx = (size_t)blockIdx.x * 256 + threadIdx.x;   // NB*NME total
    int b = (int)(idx >> 16);
    int rem = (int)(idx & 65535);
    int i = rem >> 8, j = rem & 255;
    float dm = __uint_as_float(dmaxU[b]);
    float dn = draw[idx] / dm;
    draw[idx] = dn;                                  // now holds normalized d
    float di = (float)(i - j);
    float p = expf(-di * di * (1.0f / 400.0f)) * 0.03989422804014327f; // /(delta*sqrt(2pi))
    float df = di * (1.0f / 256.0f);
    float s = 0.1f / (df * df + 1.0f);               // LAMDA1
    kmat[idx] = p * expf((s - dn) * INV_L2);
}

// ---------------------------------------------------------- Sinkhorn helpers
__device__ __forceinline__ void colpass(const float* __restrict__ kb,
                                        const float* __restrict__ in,
                                        float* __restrict__ out, int tid) {
    // out_j = sum_i k[i][j] * in_i   (coalesced: consecutive tid -> consecutive j)
    float acc = 0.f;
    for (int i = 0; i < NN; ++i) acc = fmaf(kb[(size_t)i * NN + tid], in[i], acc);
    out[tid] = acc;
}
__device__ __forceinline__ void rowpass(const float* __restrict__ kb,
                                        const float* __restrict__ in,
                                        float* __restrict__ out, int wave, int lane) {
    // out_i = 1 / (n * sum_j k[i][j] * in_j); wave-per-row, shfl reduction
    for (int it = 0; it < 32; ++it) {
        int i = it * 8 + wave;
        const float* kr = kb + (size_t)i * NN;
        float acc = 0.f;
#pragma unroll
        for (int jj = 0; jj < 8; ++jj) {
            int j = lane + jj * 32;
            acc = fmaf(kr[j], in[j], acc);
        }
        for (int m = 16; m >= 1; m >>= 1) acc += __shfl_xor(acc, m, 32);
        if (lane == 0) out[i] = 1.0f / (256.0f * acc);
    }
}

// ---------------------------------------------------------- one scan step
__global__ void k_step(const float* __restrict__ kmat, float* __restrict__ u_g,
                       float* __restrict__ v_g, float* __restrict__ crit,
                       const int* __restrict__ done, int pred) {
    if (*done) return;                       // frozen after convergence
    __shared__ float u_s[NN], z_s[NN], t2_s[NN], v2_s[NN], red[256];
    int b = blockIdx.x, tid = threadIdx.x;
    int wave = tid >> 5, lane = tid & 31;
    const float* kb = kmat + (size_t)b * NME;

    u_s[tid] = u_g[b * NN + tid];
    __syncthreads();

    // u1 = 1 / (ainvK @ (b ./ (kT @ u)))
    colpass(kb, u_s, z_s, tid); __syncthreads();
    z_s[tid] = INV_N / z_s[tid]; __syncthreads();
    rowpass(kb, z_s, u_s, wave, lane); __syncthreads();   // u_s <- u1

    if (!pred) { u_g[b * NN + tid] = u_s[tid]; return; }

    // convergence-check branch: v2 = b./(kT u1); u2 = 1/(ainvK v2)
    colpass(kb, u_s, v2_s, tid); __syncthreads();
    v2_s[tid] = INV_N / v2_s[tid]; __syncthreads();
    rowpass(kb, v2_s, t2_s, wave, lane); __syncthreads(); // t2 <- u2
    colpass(kb, t2_s, z_s, tid); __syncthreads();         // z  <- kT u2

    float cj = fabsf(v2_s[tid] * z_s[tid] - INV_N);
    red[tid] = cj; __syncthreads();
    for (int s = 128; s > 0; s >>= 1) {
        if (tid < s) red[tid] += red[tid + s];
        __syncthreads();
    }
    if (tid == 0) atomicAdd(crit, red[0]);
    u_g[b * NN + tid] = t2_s[tid];
    v_g[b * NN + tid] = v2_s[tid];
}

// ---------------------------------------------------------- fold global crit
__global__ void k_ctrl(float* crit, int* done) {
    if (threadIdx.x == 0) {
        float c = *crit;
        if (!(*done) && (c < TOLF || isnan(c))) *done = 1;
        *crit = 0.f;
    }
}

// ---------------------------------------------------------- outputs: dis, t
__global__ void k_final(const float* __restrict__ kmat, const float* __restrict__ dnorm,
                        const float* __restrict__ u_g, const float* __restrict__ v_g,
                        float* __restrict__ out) {
    __shared__ float red[256];
    __shared__ float u_s[NN], v_s[NN];
    int b = blockIdx.x, tid = threadIdx.x;
    u_s[tid] = u_g[b * NN + tid];
    v_s[tid] = v_g[b * NN + tid];
    __syncthreads();
    const float* kb = kmat + (size_t)b * NME;
    const float* db = dnorm + (size_t)b * NME;
    float* tb = out + NB + (size_t)b * NME;   // t follows dis (128 floats)
    float acc = 0.f;
    for (int c = 0; c < 256; ++c) {
        int idx = c * 256 + tid;              // row c, col tid (coalesced)
        float kv = kb[idx];
        float tv = u_s[c] * kv * v_s[tid];
        tb[idx] = tv;
        acc = fmaf(tv, db[idx], acc);         // dis = sum u_i k_ij d_ij v_j
    }
    red[tid] = acc; __syncthreads();
    for (int s = 128; s > 0; s >>= 1) {
        if (tid < s) red[tid] += red[tid + s];
        __syncthreads();
    }
    if (tid == 0) out[b] = red[0];
}

// ---------------------------------------------------------------- launcher
extern "C" void kernel_launch(void* const* d_in, const int* in_sizes, int n_in,
                              void* d_out, int out_size, void* d_ws, size_t ws_size,
                              hipStream_t stream) {
    const float* X = (const float*)d_in[0];
    const float* Y = (const float*)d_in[1];
    float* out = (float*)d_out;
    float* w = (float*)d_ws;

    const size_t NMtot = (size_t)NB * NME;     // 8,388,608
    float* dnorm = w;                          // 32 MB (raw d -> normalized d)
    float* kmat  = dnorm + NMtot;              // 32 MB
    float* u_g   = kmat + NMtot;               // NB*NN
    float* v_g   = u_g + NB * NN;
    float* shX   = v_g + NB * NN;              // NB*ND
    float* shY   = shX + NB * ND;
    float* nx    = shY + NB * ND;              // NB*NN
    float* ny    = nx + NB * NN;
    unsigned* dmaxU = (unsigned*)(ny + NB * NN);
    int*   done  = (int*)(dmaxU + NB);
    float* crit  = (float*)(done + 1);

    k_init <<<NB, 256, 0, stream>>>(u_g, v_g, dmaxU, done, crit);
    k_stats<<<NB, 256, 0, stream>>>(X, Y, shX, shY, nx, ny);
    k_gemm <<<NB * 16, 256, 0, stream>>>(X, Y, shX, shY, nx, ny, dnorm, dmaxU);
    k_expk <<<(int)(NMtot / 256), 256, 0, stream>>>(dnorm, kmat, dmaxU);

    // Host-simulated compt trajectory (deterministic until convergence; after
    // convergence the device 'done' flag freezes state, so the schedule is fixed).
    int compt = 0;
    while (compt < 200) {
        int c1 = compt + 1;
        int pred = ((c1 % 20) == 1) || (c1 == 200);
        k_step<<<NB, 256, 0, stream>>>(kmat, u_g, v_g, crit, done, pred);
        if (pred) {
            k_ctrl<<<1, 32, 0, stream>>>(crit, done);
            compt = c1 + 1;
        } else {
            compt = c1;
        }
    }

    k_final<<<NB, 256, 0, stream>>>(kmat, dnorm, u_g, v_g, out);
}